// HybridGNN_71442486002111
// MI455X (gfx1250) — compile-verified
//
#include <hip/hip_runtime.h>
#include <hip/hip_bf16.h>

// ---------------------------------------------------------------------------
// HybridGNN (GraphSAGE x2 + MLP head) for gfx1250 / MI455X.
// f32 WMMA (v_wmma_f32_16x16x4_f32) for all GEMMs; global f32 atomics for the
// segment-sum scatter (sum buffers are L2-resident: 192MB L2 >> 102MB buffer).
// ---------------------------------------------------------------------------

typedef float v2f __attribute__((ext_vector_type(2)));
typedef float v8f __attribute__((ext_vector_type(8)));

#define BM 64
#define BN 32
#define KT 16
#define GEMM_THREADS 256

// ---------------------------------------------------------------------------
__global__ void zero_f32(float* __restrict__ p, long long n) {
  long long i = (long long)blockIdx.x * blockDim.x + threadIdx.x;
  if (i < n) p[i] = 0.0f;
}

__global__ void degree_f32(const int* __restrict__ dst, float* __restrict__ cnt, int E) {
  int e = blockIdx.x * blockDim.x + threadIdx.x;
  if (e < E) atomicAdd(&cnt[dst[e]], 1.0f);
}

__global__ void rcp_cnt(float* __restrict__ cnt, int N) {
  int i = blockIdx.x * blockDim.x + threadIdx.x;
  if (i < N) cnt[i] = 1.0f / fmaxf(cnt[i], 1.0f);
}

// one thread per (edge, 4-feature chunk); lanes of a wave cover consecutive
// chunks of the same edge -> coalesced gather, scalar-broadcast src/dst loads.
__global__ void scatter_add(const float* __restrict__ feat, const int* __restrict__ src,
                            const int* __restrict__ dst, float* __restrict__ sums,
                            int E, int F) {
  const int chunks = F >> 2;
  long long tid = (long long)blockIdx.x * blockDim.x + threadIdx.x;
  long long e = tid / chunks;
  int c = (int)(tid % chunks) << 2;
  if (e >= E) return;
  int s = src[e];
  int d = dst[e];
  float4 v = *(const float4*)(feat + (long long)s * F + c);
  float* o = sums + (long long)d * F + c;
  atomicAdd(o + 0, v.x);
  atomicAdd(o + 1, v.y);
  atomicAdd(o + 2, v.z);
  atomicAdd(o + 3, v.w);
}

// ---------------------------------------------------------------------------
// out[n,j] = act( sum_k scale1[n]*A1[n,k]*B1[j,k]  +  sum_k A2[n,k]*B2[j,k] + bias[j] )
// A row-major [N,K]; B row-major [OUT,K] (torch Linear weight layout).
// act: 0=none, 1=ELU(alpha=1), 2=ReLU.  Requires K%16==0, OUT%32==0.
// Block = 256 threads = 8 waves; block tile 64(M) x 32(N); each wave: one
// 16x16 f32 WMMA accumulator, K stepped 4 at a time via wmma_f32_16x16x4_f32.
// ---------------------------------------------------------------------------
__global__ void sage_gemm(const float* __restrict__ A1, const float* __restrict__ scale1,
                          const float* __restrict__ B1,
                          const float* __restrict__ A2, const float* __restrict__ B2,
                          const float* __restrict__ bias, float* __restrict__ out,
                          int N, int K, int OUT, int act) {
  __shared__ float As[BM][KT + 1];
  __shared__ float Bs[BN][KT + 1];

  const int tid  = threadIdx.x;
  const int lane = tid & 31;
  const int wave = tid >> 5;
  const int half = lane >> 4;   // 0: lanes 0-15, 1: lanes 16-31
  const int lr   = lane & 15;
  const int m0   = (wave & 3) * 16;   // 4 M-tiles
  const int n0   = (wave >> 2) * 16;  // 2 N-tiles
  const long long rowBase = (long long)blockIdx.x * BM;
  const int colBase = blockIdx.y * BN;

  v8f acc = {0.f, 0.f, 0.f, 0.f, 0.f, 0.f, 0.f, 0.f};

  for (int phase = 0; phase < 2; ++phase) {
    const float* A  = (phase == 0) ? A1 : A2;
    const float* B  = (phase == 0) ? B1 : B2;
    const float* sc = (phase == 0) ? scale1 : nullptr;
    if (B == nullptr) continue;

    for (int k0 = 0; k0 < K; k0 += KT) {
      // stage A tile 64x16 (4 floats / thread), scaled by 1/deg for the mean
      {
        int r  = tid >> 2;
        int kk = (tid & 3) * 4;
        long long row = rowBase + r;
        float4 v = make_float4(0.f, 0.f, 0.f, 0.f);
        if (row < N) {
          v = *(const float4*)(A + row * (long long)K + k0 + kk);
          if (sc) { float s = sc[row]; v.x *= s; v.y *= s; v.z *= s; v.w *= s; }
        }
        As[r][kk + 0] = v.x; As[r][kk + 1] = v.y;
        As[r][kk + 2] = v.z; As[r][kk + 3] = v.w;
      }
      // stage B tile 32x16 (2 floats / thread)
      {
        int j  = tid >> 3;
        int kk = (tid & 7) * 2;
        const float* bp = B + (long long)(colBase + j) * K + k0 + kk;
        Bs[j][kk + 0] = bp[0];
        Bs[j][kk + 1] = bp[1];
      }
      __syncthreads();

#pragma unroll
      for (int kk = 0; kk < KT; kk += 4) {
        // A frag: lane holds (M=lr, K=kk+2*half+v); B frag mirrored on N.
        v2f a, b;
        a.x = As[m0 + lr][kk + 2 * half + 0];
        a.y = As[m0 + lr][kk + 2 * half + 1];
        b.x = Bs[n0 + lr][kk + 2 * half + 0];
        b.y = Bs[n0 + lr][kk + 2 * half + 1];
        acc = __builtin_amdgcn_wmma_f32_16x16x4_f32(
            /*neg_a=*/false, a, /*neg_b=*/false, b,
            /*c_mod=*/(short)0, acc, /*reuse_a=*/false, /*reuse_b=*/false);
      }
      __syncthreads();
    }
  }

  // epilogue: C VGPR r -> row m0 + r + 8*half, col n0 + lr
  const int col = colBase + n0 + lr;
  const float bb = bias ? bias[col] : 0.0f;
#pragma unroll
  for (int r = 0; r < 8; ++r) {
    long long row = rowBase + m0 + r + 8 * half;
    if (row < N) {
      float v = acc[r] + bb;
      if (act == 1)      v = (v > 0.f) ? v : expm1f(v);
      else if (act == 2) v = fmaxf(v, 0.f);
      out[row * (long long)OUT + col] = v;
    }
  }
}

// final 128 -> 1 projection: one wave32 per node, shuffle reduction
__global__ void head_dot(const float* __restrict__ h, const float* __restrict__ w,
                         const float* __restrict__ b, float* __restrict__ out, int N) {
  int wave = threadIdx.x >> 5;
  int lane = threadIdx.x & 31;
  long long n = (long long)blockIdx.x * 8 + wave;
  if (n >= N) return;
  const float* hp = h + n * 128;
  float s = hp[lane]      * w[lane]
          + hp[lane + 32] * w[lane + 32]
          + hp[lane + 64] * w[lane + 64]
          + hp[lane + 96] * w[lane + 96];
#pragma unroll
  for (int off = 16; off > 0; off >>= 1) s += __shfl_xor(s, off, 32);
  if (lane == 0) out[n] = s + b[0];
}

// ---------------------------------------------------------------------------
extern "C" void kernel_launch(void* const* d_in, const int* in_sizes, int n_in,
                              void* d_out, int out_size, void* d_ws, size_t ws_size,
                              hipStream_t stream) {
  const float* x   = (const float*)d_in[0];
  const int*   ei  = (const int*)  d_in[1];
  const float* W1l = (const float*)d_in[2];
  const float* b1  = (const float*)d_in[3];
  const float* W1r = (const float*)d_in[4];
  const float* W2l = (const float*)d_in[5];
  const float* b2  = (const float*)d_in[6];
  const float* W2r = (const float*)d_in[7];
  const float* Wf1 = (const float*)d_in[8];
  const float* bf1 = (const float*)d_in[9];
  const float* Wf2 = (const float*)d_in[10];
  const float* bf2 = (const float*)d_in[11];
  float* out = (float*)d_out;
  (void)n_in; (void)out_size; (void)ws_size;

  const int IN_C = 128, HID = 256;
  const int N = in_sizes[0] / IN_C;
  const int E = in_sizes[1] / 2;
  const int* src = ei;
  const int* dst = ei + E;

  // carve workspace (256B aligned chunks)
  char* wsp = (char*)d_ws;
  auto carve = [&](size_t bytes) -> float* {
    float* p = (float*)wsp;
    wsp += (bytes + 255) & ~(size_t)255;
    return p;
  };
  float* cnt  = carve((size_t)N * 4);             // reciprocal degree
  float* sums = carve((size_t)N * HID * 4);       // segment sums (both layers)
  float* h1   = carve((size_t)N * HID * 4);
  float* h2   = carve((size_t)N * HID * 4);
  float* h3   = carve((size_t)N * (HID / 2) * 4);

  // --- degree + reciprocal ---
  zero_f32<<<(N + 255) / 256, 256, 0, stream>>>(cnt, (long long)N);
  degree_f32<<<(E + 255) / 256, 256, 0, stream>>>(dst, cnt, E);
  rcp_cnt<<<(N + 255) / 256, 256, 0, stream>>>(cnt, N);

  // --- layer 1: scatter x -> sums, fused dual GEMM + ELU ---
  {
    long long nz = (long long)N * IN_C;
    zero_f32<<<(int)((nz + 255) / 256), 256, 0, stream>>>(sums, nz);
    long long t = (long long)E * (IN_C / 4);
    scatter_add<<<(int)((t + 255) / 256), 256, 0, stream>>>(x, src, dst, sums, E, IN_C);
    dim3 grid((N + BM - 1) / BM, HID / BN);
    sage_gemm<<<grid, GEMM_THREADS, 0, stream>>>(sums, cnt, W1l, x, W1r, b1, h1,
                                                 N, IN_C, HID, /*ELU*/1);
  }

  // --- layer 2: scatter h1 -> sums, fused dual GEMM + ELU ---
  {
    long long nz = (long long)N * HID;
    zero_f32<<<(int)((nz + 255) / 256), 256, 0, stream>>>(sums, nz);
    long long t = (long long)E * (HID / 4);
    scatter_add<<<(int)((t + 255) / 256), 256, 0, stream>>>(h1, src, dst, sums, E, HID);
    dim3 grid((N + BM - 1) / BM, HID / BN);
    sage_gemm<<<grid, GEMM_THREADS, 0, stream>>>(sums, cnt, W2l, h1, W2r, b2, h2,
                                                 N, HID, HID, /*ELU*/1);
  }

  // --- MLP head: ReLU(h2 @ Wf1.T + bf1) then dot with Wf2 ---
  {
    dim3 grid((N + BM - 1) / BM, (HID / 2) / BN);
    sage_gemm<<<grid, GEMM_THREADS, 0, stream>>>(h2, nullptr, Wf1, nullptr, nullptr,
                                                 bf1, h3, N, HID, HID / 2, /*ReLU*/2);
  }
  head_dot<<<(N + 7) / 8, 256, 0, stream>>>(h3, Wf2, bf2, out, N);
}